// Encoder_34522947125909
// MI455X (gfx1250) — compile-verified
//
#include <hip/hip_runtime.h>
#include <hip/hip_bf16.h>
#include <math.h>

// ---------------------------------------------------------------------------
// Bidirectional 2-layer GRU encoder on CDNA5 (gfx1250), fp32 via
// V_WMMA_F32_16X16X4_F32.  B=4096, T=72, I=9, U=64, gate order z,r,h,
// Keras reset_after=True semantics.
// ---------------------------------------------------------------------------

typedef float v2f __attribute__((ext_vector_type(2)));
typedef float v8f __attribute__((ext_vector_type(8)));

#define T_LEN 72

__device__ __forceinline__ v8f wmma4(v2f a, v2f b, v8f c) {
  // D = A(16x4) x B(4x16) + C(16x16), all fp32
  return __builtin_amdgcn_wmma_f32_16x16x4_f32(
      /*neg_a=*/false, a, /*neg_b=*/false, b,
      /*c_mod=*/(short)0, c, /*reuse_a=*/false, /*reuse_b=*/false);
}

__device__ __forceinline__ v8f splat8(float s) {
  v8f v = {s, s, s, s, s, s, s, s};
  return v;
}

__device__ __forceinline__ float sigmoid_f(float x) {
  return 1.0f / (1.0f + __expf(-x));
}
__device__ __forceinline__ float tanh_f(float x) {
  float e = __expf(2.0f * x);
  return (e - 1.0f) / (e + 1.0f);
}

// Pre-swizzle a row-major weight matrix W[Ktot x 192] into WMMA B-fragment
// order: frag (kt,nt) -> 32 lanes x 2 floats, lane<16 holds K=4kt+{0,1},
// lane>=16 holds K=4kt+2+{0,1}; N = nt*16 + (lane&15).  Zero-pads K beyond
// Ktot (used for I=9 -> 12).
__device__ __forceinline__ void fill_wfrag(float* dst, const float* __restrict__ W,
                                           int KT, int Ktot, int tid, int nthreads) {
  int total = KT * 12 * 64;
  for (int e = tid; e < total; e += nthreads) {
    int f = e >> 6;         // frag index kt*12+nt
    int r = e & 63;         // lane*2 + p
    int laneF = r >> 1, p = r & 1;
    int kt = f / 12, nt = f - kt * 12;
    int hiF = laneF >> 4, loF = laneF & 15;
    int k = 4 * kt + 2 * hiF + p;
    int n = nt * 16 + loF;
    dst[e] = (k < Ktot) ? W[k * 192 + n] : 0.0f;
  }
}

// ---------------------------------------------------------------------------
// Kernel 1: layer e1 (bidirectional, relu).  4 waves/WG:
//   wave = dir (f/b) x batch-tile (2 tiles of 16 rows).  Writes h1[B,T,128].
// ---------------------------------------------------------------------------
__global__ void __launch_bounds__(128, 1) gru_e1_kernel(
    const float* __restrict__ x,
    const float* __restrict__ Wf, const float* __restrict__ Uf, const float* __restrict__ bf,
    const float* __restrict__ Wb, const float* __restrict__ Ub, const float* __restrict__ bb,
    float* __restrict__ h1, int B)
{
  extern __shared__ float smem[];
  float* lWf = smem;              // 3*12*64   = 2304
  float* lWb = smem + 2304;       //             2304
  float* lUf = smem + 4608;       // 16*12*64  = 12288
  float* lUb = smem + 16896;      //             12288
  float* lx  = smem + 29184;      // [32][72][12] = 27648 (K padded 9->12)
  float* lhs = smem + 56832;      // 4 waves * 16*68 = 4352

  const int tid  = threadIdx.x;
  const int wave = tid >> 5, lane = tid & 31;
  const int dir  = wave & 1, tile = wave >> 1;
  const int wgb  = blockIdx.x * 32;
  const int b0   = wgb + tile * 16;
  const int hi = lane >> 4, lo = lane & 15;

  fill_wfrag(lWf, Wf, 3, 9, tid, 128);
  fill_wfrag(lWb, Wb, 3, 9, tid, 128);
  fill_wfrag(lUf, Uf, 16, 64, tid, 128);
  fill_wfrag(lUb, Ub, 16, 64, tid, 128);
  for (int e = tid; e < 32 * 72 * 12; e += 128) {
    int m = e / 864, rem = e - m * 864;
    int t = rem / 12, c = rem - t * 12;
    lx[e] = (c < 9) ? x[((size_t)(wgb + m) * T_LEN + t) * 9 + c] : 0.0f;
  }
  for (int e = tid; e < 4 * 1088; e += 128) lhs[e] = 0.0f;   // h0 = 0
  __syncthreads();

  const float* lU   = dir ? lUb : lUf;
  const float* lW   = dir ? lWb : lWf;
  const float* bias = dir ? bb  : bf;
  float* hs = lhs + wave * 1088;            // [16][68] h scratch (D->A relayout)
  const float* xrow = lx + tile * 13824;    // this tile's [16][72][12]

  // bias pre-splats: z,r tiles fuse b0+b1; h tile keeps rec(b1) and x(b0) apart
  float binit[12], bxinit[4];
  #pragma unroll
  for (int n = 0; n < 12; ++n) {
    float v = bias[192 + n * 16 + lo];
    if (n < 8) v += bias[n * 16 + lo];
    binit[n] = v;
  }
  #pragma unroll
  for (int n = 0; n < 4; ++n) bxinit[n] = bias[(8 + n) * 16 + lo];

  v8f hD[4];
  #pragma unroll
  for (int n = 0; n < 4; ++n) hD[n] = splat8(0.0f);

  for (int t = 0; t < T_LEN; ++t) {
    const int tx = dir ? (T_LEN - 1 - t) : t;
    v8f acc[12], accx[4];
    #pragma unroll
    for (int n = 0; n < 12; ++n) acc[n] = splat8(binit[n]);
    #pragma unroll
    for (int n = 0; n < 4; ++n)  accx[n] = splat8(bxinit[n]);

    // recurrent: h[16x64] @ U[64x192]  (16 K-tiles x 12 N-tiles of WMMA)
    for (int kt = 0; kt < 16; ++kt) {
      v2f a = *(const v2f*)(hs + lo * 68 + 4 * kt + 2 * hi);
      #pragma unroll
      for (int nt = 0; nt < 12; ++nt) {
        v2f bm = *(const v2f*)(lU + ((kt * 12 + nt) * 32 + lane) * 2);
        acc[nt] = wmma4(a, bm, acc[nt]);
      }
    }
    // input projection: x_t[16x12] @ W[12x192]; h-gate chunk kept separate
    #pragma unroll
    for (int kt = 0; kt < 3; ++kt) {
      v2f a = *(const v2f*)(xrow + lo * 864 + tx * 12 + 4 * kt + 2 * hi);
      #pragma unroll
      for (int nt = 0; nt < 8; ++nt) {
        v2f bm = *(const v2f*)(lW + ((kt * 12 + nt) * 32 + lane) * 2);
        acc[nt] = wmma4(a, bm, acc[nt]);
      }
      #pragma unroll
      for (int nt = 8; nt < 12; ++nt) {
        v2f bm = *(const v2f*)(lW + ((kt * 12 + nt) * 32 + lane) * 2);
        accx[nt - 8] = wmma4(a, bm, accx[nt - 8]);
      }
    }
    // gates (activation = relu); lane-aligned across tiles in D layout
    #pragma unroll
    for (int n = 0; n < 4; ++n) {
      #pragma unroll
      for (int j = 0; j < 8; ++j) {
        float z  = sigmoid_f(acc[n][j]);
        float r  = sigmoid_f(acc[n + 4][j]);
        float hh = fmaxf(accx[n][j] + r * acc[n + 8][j], 0.0f);
        float hn = z * hD[n][j] + (1.0f - z) * hh;
        hD[n][j] = hn;
        hs[(j + 8 * hi) * 68 + n * 16 + lo] = hn;
        h1[((size_t)(b0 + j + 8 * hi) * T_LEN + tx) * 128 + dir * 64 + n * 16 + lo] = hn;
      }
    }
    __syncthreads();   // make D-layout stores visible to next step's A loads
  }
}

// ---------------------------------------------------------------------------
// Kernel 2: layer e2 (tanh), one direction per launch.  4 waves/WG, each wave
// owns a 16-row batch tile.  Reads h1[B,T,128]; writes last-state outputs.
// ---------------------------------------------------------------------------
__global__ void __launch_bounds__(128, 1) gru_e2_kernel(
    const float* __restrict__ h1,
    const float* __restrict__ W, const float* __restrict__ U, const float* __restrict__ bias,
    float* __restrict__ out, int B, int dir)
{
  extern __shared__ float smem[];
  float* lW    = smem;            // 32*12*64 = 24576
  float* lU    = smem + 24576;    // 16*12*64 = 12288
  float* lhs   = smem + 36864;    // 4 waves * 16*68  = 4352
  float* ltile = smem + 41216;    // 4 waves * 16*132 = 8448 (h1[t] staging)

  const int tid  = threadIdx.x;
  const int wave = tid >> 5, lane = tid & 31;
  const int b0   = blockIdx.x * 64 + wave * 16;
  const int hi = lane >> 4, lo = lane & 15;

  fill_wfrag(lW, W, 32, 128, tid, 128);
  fill_wfrag(lU, U, 16, 64, tid, 128);
  for (int e = tid; e < 4 * 1088; e += 128) lhs[e] = 0.0f;
  __syncthreads();

  float* hs    = lhs + wave * 1088;
  float* tilep = ltile + wave * 2112;

  float binit[12], bxinit[4];
  #pragma unroll
  for (int n = 0; n < 12; ++n) {
    float v = bias[192 + n * 16 + lo];
    if (n < 8) v += bias[n * 16 + lo];
    binit[n] = v;
  }
  #pragma unroll
  for (int n = 0; n < 4; ++n) bxinit[n] = bias[(8 + n) * 16 + lo];

  v8f hD[4];
  #pragma unroll
  for (int n = 0; n < 4; ++n) hD[n] = splat8(0.0f);

  for (int t = 0; t < T_LEN; ++t) {
    const int tx = dir ? (T_LEN - 1 - t) : t;

    // stage this wave's h1[:, tx, :] tile (16 x 128) into LDS, float4 wide
    for (int e = lane; e < 512; e += 32) {
      int m = e >> 5, q = e & 31;
      float4 v = *(const float4*)(h1 + ((size_t)(b0 + m) * T_LEN + tx) * 128 + q * 4);
      *(float4*)(tilep + m * 132 + q * 4) = v;
    }
    __syncthreads();

    v8f acc[12], accx[4];
    #pragma unroll
    for (int n = 0; n < 12; ++n) acc[n] = splat8(binit[n]);
    #pragma unroll
    for (int n = 0; n < 4; ++n)  accx[n] = splat8(bxinit[n]);

    // recurrent: h[16x64] @ U[64x192]
    for (int kt = 0; kt < 16; ++kt) {
      v2f a = *(const v2f*)(hs + lo * 68 + 4 * kt + 2 * hi);
      #pragma unroll
      for (int nt = 0; nt < 12; ++nt) {
        v2f bm = *(const v2f*)(lU + ((kt * 12 + nt) * 32 + lane) * 2);
        acc[nt] = wmma4(a, bm, acc[nt]);
      }
    }
    // input projection: h1_t[16x128] @ W[128x192]
    for (int kt = 0; kt < 32; ++kt) {
      v2f a = *(const v2f*)(tilep + lo * 132 + 4 * kt + 2 * hi);
      #pragma unroll
      for (int nt = 0; nt < 8; ++nt) {
        v2f bm = *(const v2f*)(lW + ((kt * 12 + nt) * 32 + lane) * 2);
        acc[nt] = wmma4(a, bm, acc[nt]);
      }
      #pragma unroll
      for (int nt = 8; nt < 12; ++nt) {
        v2f bm = *(const v2f*)(lW + ((kt * 12 + nt) * 32 + lane) * 2);
        accx[nt - 8] = wmma4(a, bm, accx[nt - 8]);
      }
    }
    // gates (activation = tanh)
    #pragma unroll
    for (int n = 0; n < 4; ++n) {
      #pragma unroll
      for (int j = 0; j < 8; ++j) {
        float z  = sigmoid_f(acc[n][j]);
        float r  = sigmoid_f(acc[n + 4][j]);
        float hh = tanh_f(accx[n][j] + r * acc[n + 8][j]);
        float hn = z * hD[n][j] + (1.0f - z) * hh;
        hD[n][j] = hn;
        hs[(j + 8 * hi) * 68 + n * 16 + lo] = hn;
      }
    }
    __syncthreads();
  }

  // final states -> d_out: [B,128] concat, then f2 [B,64], then b2 [B,64]
  const size_t o2 = (size_t)B * 128 + (size_t)dir * (size_t)B * 64;
  #pragma unroll
  for (int n = 0; n < 4; ++n) {
    #pragma unroll
    for (int j = 0; j < 8; ++j) {
      int m = b0 + j + 8 * hi;
      int u = n * 16 + lo;
      float v = hD[n][j];
      out[(size_t)m * 128 + dir * 64 + u] = v;
      out[o2 + (size_t)m * 64 + u] = v;
    }
  }
}

// ---------------------------------------------------------------------------
extern "C" void kernel_launch(void* const* d_in, const int* in_sizes, int n_in,
                              void* d_out, int out_size, void* d_ws, size_t ws_size,
                              hipStream_t stream) {
  const float* x    = (const float*)d_in[0];
  const float* e1fW = (const float*)d_in[1];
  const float* e1fU = (const float*)d_in[2];
  const float* e1fb = (const float*)d_in[3];
  const float* e1bW = (const float*)d_in[4];
  const float* e1bU = (const float*)d_in[5];
  const float* e1bb = (const float*)d_in[6];
  const float* e2fW = (const float*)d_in[7];
  const float* e2fU = (const float*)d_in[8];
  const float* e2fb = (const float*)d_in[9];
  const float* e2bW = (const float*)d_in[10];
  const float* e2bU = (const float*)d_in[11];
  const float* e2bb = (const float*)d_in[12];
  float* out = (float*)d_out;
  float* h1  = (float*)d_ws;                 // [B,72,128] fp32 intermediate
  const int B = in_sizes[0] / (T_LEN * 9);

  const size_t lds1 = 61184 * sizeof(float); // ~239 KB
  const size_t lds2 = 49664 * sizeof(float); // ~194 KB

  hipLaunchKernelGGL(gru_e1_kernel, dim3(B / 32), dim3(128), lds1, stream,
                     x, e1fW, e1fU, e1fb, e1bW, e1bU, e1bb, h1, B);
  hipLaunchKernelGGL(gru_e2_kernel, dim3(B / 64), dim3(128), lds2, stream,
                     h1, e2fW, e2fU, e2fb, out, B, 0);
  hipLaunchKernelGGL(gru_e2_kernel, dim3(B / 64), dim3(128), lds2, stream,
                     h1, e2bW, e2bU, e2bb, out, B, 1);
}